// ImprovedGNN_84232898609317
// MI455X (gfx1250) — compile-verified
//
#include <hip/hip_runtime.h>
#include <hip/hip_bf16.h>

typedef float v2f __attribute__((ext_vector_type(2)));
typedef float v8f __attribute__((ext_vector_type(8)));

#define N_NODES 50000
#define N_EDGES 800000
#define HIDDEN  128
#define N_LAYERS 3
#define EPS_BN  1e-5f
#define NH (N_NODES * HIDDEN)

// ---------------------------------------------------------------- utilities
__global__ void zero_f32(float* __restrict__ p, int n) {
    int i = blockIdx.x * blockDim.x + threadIdx.x;
    if (i < n) p[i] = 0.0f;
}

__global__ void degree_kernel(const long long* __restrict__ dst, float* __restrict__ deg) {
    int e = blockIdx.x * blockDim.x + threadIdx.x;
    if (e < N_EDGES) atomicAdd(&deg[(int)dst[e]], 1.0f);
}

__global__ void invdeg_kernel(float* __restrict__ deg) {
    int i = blockIdx.x * blockDim.x + threadIdx.x;
    if (i < N_NODES) deg[i] = 1.0f / fmaxf(deg[i], 1.0f);
}

// ------------------------------------------------------- edge scatter (mean)
// One wave32 per edge: each lane moves 4 consecutive channels (float4 gather,
// 4x global_atomic_add_f32 scatter). 512B coalesced per edge row.
__global__ __launch_bounds__(256) void scatter_kernel(
    const long long* __restrict__ src, const long long* __restrict__ dst,
    const float* __restrict__ x, float* __restrict__ agg)
{
    int wave = (blockIdx.x * blockDim.x + threadIdx.x) >> 5;
    int lane = threadIdx.x & 31;
    if (wave >= N_EDGES) return;
    int s = (int)src[wave];
    int d = (int)dst[wave];
    const float4 v = ((const float4*)(x + (size_t)s * HIDDEN))[lane];
    float* ap = agg + (size_t)d * HIDDEN + lane * 4;
    atomicAdd(ap + 0, v.x);
    atomicAdd(ap + 1, v.y);
    atomicAdd(ap + 2, v.z);
    atomicAdd(ap + 3, v.w);
}

// ------------------------------------------------- fused dual-GEMM + stats
// h = (agg*invdeg) @ Wl^T + bl + x @ Wr^T, per-channel sum/sumsq for BN.
// One wave computes a 16x128 output strip via V_WMMA_F32_16X16X4_F32.
__global__ __launch_bounds__(256) void gemm_bias_stats(
    const float* __restrict__ agg, const float* __restrict__ xin,
    const float* __restrict__ Wl,  const float* __restrict__ Wr,
    const float* __restrict__ bl,  const float* __restrict__ invdeg,
    float* __restrict__ h, float* __restrict__ gsum, float* __restrict__ gsq)
{
    __shared__ float lds_sum[HIDDEN];
    __shared__ float lds_sq[HIDDEN];
    const int tid = threadIdx.x;
    if (tid < HIDDEN) { lds_sum[tid] = 0.0f; lds_sq[tid] = 0.0f; }
    __syncthreads();

    const int wave  = tid >> 5;
    const int lane  = tid & 31;
    const int lhalf = lane >> 4;          // 0: K lo pair / M 0..7 ; 1: K hi pair / M 8..15
    const int lmod  = lane & 15;
    const int r0    = (blockIdx.x * 8 + wave) * 16;   // first row of this wave's strip
    const int aRow  = r0 + lmod;                      // A-fragment row for this lane
    const int mBase = r0 + lhalf * 8;                 // rows covered by acc components

    v8f acc[8] = {};

    const int aRowC = (aRow < N_NODES) ? aRow : (N_NODES - 1);   // clamped (masked via scale)

    #pragma unroll 1
    for (int pass = 0; pass < 2; ++pass) {
        const float* A = pass ? xin : agg;
        const float* W = pass ? Wr : Wl;
        float scale;
        if (aRow < N_NODES) scale = pass ? 1.0f : invdeg[aRow];
        else                scale = 0.0f;     // zero out-of-range rows -> exact stats
        const float* aBase = A + (size_t)aRowC * HIDDEN + 2 * lhalf;

        #pragma unroll 1
        for (int k0 = 0; k0 < HIDDEN; k0 += 4) {
            v2f a = *(const v2f*)(aBase + k0);
            a *= scale;
            #pragma unroll
            for (int nt = 0; nt < 8; ++nt) {
                const int n = nt * 16 + lmod;
                v2f b = *(const v2f*)(W + (size_t)n * HIDDEN + k0 + 2 * lhalf);
                acc[nt] = __builtin_amdgcn_wmma_f32_16x16x4_f32(
                    false, a, false, b, (short)0, acc[nt], false, false);
            }
        }
    }

    // epilogue: add bias, store h, accumulate per-channel stats
    if (r0 + 16 <= N_NODES) {
        // Fast path (wave-uniform): all 16 rows valid -> no exec-mask churn.
        #pragma unroll
        for (int nt = 0; nt < 8; ++nt) {
            const int n = nt * 16 + lmod;
            const float bias = bl[n];
            float* hp = h + (size_t)mBase * HIDDEN + n;
            float s1 = 0.0f, s2 = 0.0f;
            #pragma unroll
            for (int v = 0; v < 8; ++v) {
                const float val = acc[nt][v] + bias;
                hp[(size_t)v * HIDDEN] = val;
                s1 += val;
                s2 += val * val;
            }
            atomicAdd(&lds_sum[n], s1);   // ds_add_f32
            atomicAdd(&lds_sq[n],  s2);
        }
    } else {
        // Tail block: per-row guard keeps stats exact.
        #pragma unroll
        for (int nt = 0; nt < 8; ++nt) {
            const int n = nt * 16 + lmod;
            const float bias = bl[n];
            float s1 = 0.0f, s2 = 0.0f;
            #pragma unroll
            for (int v = 0; v < 8; ++v) {
                const int r = mBase + v;
                if (r < N_NODES) {
                    const float val = acc[nt][v] + bias;
                    h[(size_t)r * HIDDEN + n] = val;
                    s1 += val;
                    s2 += val * val;
                }
            }
            atomicAdd(&lds_sum[n], s1);
            atomicAdd(&lds_sq[n],  s2);
        }
    }
    __syncthreads();
    if (tid < HIDDEN)            atomicAdd(&gsum[tid], lds_sum[tid]);
    else if (tid < 2 * HIDDEN)   atomicAdd(&gsq[tid - HIDDEN], lds_sq[tid - HIDDEN]);
}

// ------------------------------------------------------------- BN finalize
__global__ void finalize_stats(const float* __restrict__ gsum, const float* __restrict__ gsq,
                               float* __restrict__ mean, float* __restrict__ rstd)
{
    int c = threadIdx.x;
    if (c < HIDDEN) {
        const float m = gsum[c] * (1.0f / (float)N_NODES);
        const float v = gsq[c] * (1.0f / (float)N_NODES) - m * m;
        mean[c] = m;
        rstd[c] = rsqrtf(v + EPS_BN);
    }
}

// ------------------------------------------- BN apply + ReLU + residual
__global__ __launch_bounds__(256) void bn_relu_res(
    const float* __restrict__ h, const float* __restrict__ xres,
    const float* __restrict__ gamma, const float* __restrict__ beta,
    const float* __restrict__ mean,  const float* __restrict__ rstd,
    float* __restrict__ out, int addRes)
{
    const int idx = blockIdx.x * blockDim.x + threadIdx.x;   // over NH/4
    if (idx >= NH / 4) return;
    const int cg = idx & (HIDDEN / 4 - 1);                   // channel group (4 ch)
    const float4 g = ((const float4*)gamma)[cg];
    const float4 b = ((const float4*)beta)[cg];
    const float4 m = ((const float4*)mean)[cg];
    const float4 r = ((const float4*)rstd)[cg];
    const float4 hv = ((const float4*)h)[idx];
    float4 o;
    o.x = fmaxf(g.x * (hv.x - m.x) * r.x + b.x, 0.0f);
    o.y = fmaxf(g.y * (hv.y - m.y) * r.y + b.y, 0.0f);
    o.z = fmaxf(g.z * (hv.z - m.z) * r.z + b.z, 0.0f);
    o.w = fmaxf(g.w * (hv.w - m.w) * r.w + b.w, 0.0f);
    if (addRes) {
        const float4 xv = ((const float4*)xres)[idx];
        o.x += xv.x; o.y += xv.y; o.z += xv.z; o.w += xv.w;
    }
    ((float4*)out)[idx] = o;
}

// ----------------------------------------------------------------- launcher
extern "C" void kernel_launch(void* const* d_in, const int* in_sizes, int n_in,
                              void* d_out, int out_size, void* d_ws, size_t ws_size,
                              hipStream_t stream) {
    const float*     x_in = (const float*)d_in[0];
    const long long* ei   = (const long long*)d_in[1];   // int64 [2, E]
    const float*     Wl   = (const float*)d_in[2];
    const float*     bl   = (const float*)d_in[3];
    const float*     Wr   = (const float*)d_in[4];
    const float*     gamma= (const float*)d_in[5];
    const float*     beta = (const float*)d_in[6];
    float* out = (float*)d_out;

    const long long* src = ei;
    const long long* dst = ei + N_EDGES;

    float* ws   = (float*)d_ws;
    float* agg  = ws;                 // NH floats; also aliased as h
    float* xmid = ws + (size_t)NH;    // NH floats (layer-1 output)
    float* deg  = ws + (size_t)2 * NH;        // N_NODES floats -> becomes invdeg
    float* gsum = deg + 50048;        // 128
    float* gsq  = gsum + HIDDEN;      // 128
    float* mean = gsq + HIDDEN;       // 128
    float* rstd = mean + HIDDEN;      // 128

    // degree -> inv_deg (recomputed every call: deterministic)
    zero_f32<<<(N_NODES + 255) / 256, 256, 0, stream>>>(deg, N_NODES);
    degree_kernel<<<(N_EDGES + 255) / 256, 256, 0, stream>>>(dst, deg);
    invdeg_kernel<<<(N_NODES + 255) / 256, 256, 0, stream>>>(deg);

    for (int i = 0; i < N_LAYERS; ++i) {
        const float* x_cur  = (i == 0) ? x_in : ((i == 1) ? out : xmid);
        float*       x_next = (i == 1) ? xmid : out;

        zero_f32<<<(NH + 255) / 256, 256, 0, stream>>>(agg, NH);
        zero_f32<<<1, 256, 0, stream>>>(gsum, 2 * HIDDEN);

        scatter_kernel<<<N_EDGES / 8, 256, 0, stream>>>(src, dst, x_cur, agg);

        gemm_bias_stats<<<(N_NODES + 127) / 128, 256, 0, stream>>>(
            agg, x_cur,
            Wl + (size_t)i * HIDDEN * HIDDEN, Wr + (size_t)i * HIDDEN * HIDDEN,
            bl + (size_t)i * HIDDEN, deg,
            agg /* h aliases agg: each wave writes only rows it read */,
            gsum, gsq);

        finalize_stats<<<1, HIDDEN, 0, stream>>>(gsum, gsq, mean, rstd);

        bn_relu_res<<<(NH / 4 + 255) / 256, 256, 0, stream>>>(
            agg, x_cur,
            gamma + (size_t)i * HIDDEN, beta + (size_t)i * HIDDEN,
            mean, rstd, x_next, (i > 0) ? 1 : 0);
    }
}